// Triplet_MMD_62251255988388
// MI455X (gfx1250) — compile-verified
//
#include <hip/hip_runtime.h>
#include <math.h>

typedef __attribute__((ext_vector_type(2))) float v2f;
typedef __attribute__((ext_vector_type(8))) float v8f;

#define TT   512
#define DF   128
#define NW   16
#define STEPM1 15            // STEP-1
#define PAIRS 240
#define CELLS 262144         // 512*512 cells per pair (diag-major, unpadded)
#define INFV 1e10f

// ---------------------------------------------------------------------------
// diag-major addressing: cell (m,n), d = m+n in [0,1022]
//   off(d) = d<=511 ? d(d+1)/2 : 262144 - (1023-d)(1024-d)/2
//   index  = off(d) + m - max(0, d-511)
// consecutive m on a diagonal are contiguous -> coalesced DP loads.
// ---------------------------------------------------------------------------
__device__ __forceinline__ long diag_off(int d) {
    return (d <= 511) ? (long)d * (d + 1) / 2
                      : (long)CELLS - (long)(1023 - d) * (1024 - d) / 2;
}

// ---------------------- kernel 0: per-row squared norms --------------------
// one wave per (seq, t) row of 128 floats; 131072 rows
__global__ void norms_kernel(const float* __restrict__ data, float* __restrict__ norms) {
    int wid  = (blockIdx.x * blockDim.x + threadIdx.x) >> 5;
    int lane = threadIdx.x & 31;
    const float4* p = (const float4*)(data + (size_t)wid * DF);
    float4 v = p[lane];
    float s = v.x * v.x + v.y * v.y + v.z * v.z + v.w * v.w;
    #pragma unroll
    for (int o = 16; o; o >>= 1) s += __shfl_xor(s, o, 32);
    if (lane == 0) norms[wid] = s;
}

// ---------------------- kernel 1: WMMA fp32 GEMM + epilogue ----------------
// grid = nPairs*256 blocks of 128 threads (4 waves). Each wave -> one 16x16
// tile of D for one pair. 4 tiles in a block share trow -> shared A stage.
__global__ void gemm_kernel(const float* __restrict__ data,
                            const float* __restrict__ norms,
                            float* __restrict__ Dd, int chunk0) {
    __shared__ float Xs[16 * DF];        // 8 KiB, shared by all 4 waves
    __shared__ float Ys[4][16 * DF];     // 8 KiB per wave

    int block_tile0 = blockIdx.x * 4;
    int pair_local  = block_tile0 >> 10;          // 1024 tiles per pair
    int tl0         = block_tile0 & 1023;
    int trow        = tl0 >> 5;
    int pair        = chunk0 + pair_local;
    int wrow        = pair / STEPM1;
    int rcol        = pair % STEPM1;
    int aseq        = wrow * (STEPM1 + 1);
    int oseq        = aseq + 1 + rcol;
    const float* X  = data + (size_t)aseq * TT * DF;
    const float* Y  = data + (size_t)oseq * TT * DF;

    int row0 = trow * 16;
    int wv   = threadIdx.x >> 5;
    int lane = threadIdx.x & 31;
    int col0 = ((tl0 & 31) + wv) * 16;

    // stage A rows [row0, row0+16) cooperatively (coalesced float4)
    {
        const float4* src = (const float4*)(X + (size_t)row0 * DF);
        float4* dst = (float4*)Xs;
        for (int i = threadIdx.x; i < 16 * DF / 4; i += 128) dst[i] = src[i];
    }
    // each wave stages its own B rows [col0, col0+16)
    {
        const float4* src = (const float4*)(Y + (size_t)col0 * DF);
        float4* dst = (float4*)Ys[wv];
        for (int i = lane; i < 16 * DF / 4; i += 32) dst[i] = src[i];
    }
    __syncthreads();

    int lm = lane & 15, hi = lane >> 4;
    const float* xr = Xs + lm * DF;
    const float* yr = Ys[wv] + lm * DF;

    v8f acc = {};
    #pragma unroll 4
    for (int k0 = 0; k0 < DF; k0 += 4) {
        // f32 A 16x4: lanes 0-15 M=0..15 {K=k0,k0+1}; lanes 16-31 {K=k0+2,k0+3}
        v2f a = *(const v2f*)(xr + k0 + 2 * hi);
        v2f b = *(const v2f*)(yr + k0 + 2 * hi);
        acc = __builtin_amdgcn_wmma_f32_16x16x4_f32(false, a, false, b,
                                                    (short)0, acc, false, false);
    }

    // epilogue: D = ||x||^2 + ||y||^2 - 2 s ; scatter into diag-major layout.
    // index advanced incrementally: (m,d)->(m+1,d+1) is += len(d)+1-(d>=511).
    float* Dp = Dd + (size_t)pair_local * CELLS;
    int n  = col0 + lm;
    float yn = norms[(size_t)oseq * TT + n];
    int m0 = row0 + 8 * hi;                        // C layout: vgpr r, halves +8
    int d0 = m0 + n;
    int mmin0 = d0 > 511 ? d0 - 511 : 0;
    long idx = diag_off(d0) + (m0 - mmin0);
    #pragma unroll
    for (int r = 0; r < 8; ++r) {
        int m = m0 + r;
        float xn = norms[(size_t)aseq * TT + m];
        Dp[idx] = xn + yn - 2.0f * acc[r];
        int d = d0 + r;
        int len = (d <= 511) ? (d + 1) : (1023 - d);
        idx += len + 1 - (d >= 511 ? 1 : 0);
    }
}

// ---------------------- kernel 2: soft-DTW anti-diagonal DP ----------------
// one workgroup (512 threads) per pair; 3 rotating diagonals in LDS.
// Diagonal base pointer advanced incrementally (scalar add per step).
__global__ void dtw_kernel(const float* __restrict__ Dd,
                           const int* __restrict__ lens,
                           float* __restrict__ dists, int chunk0) {
    __shared__ float buf[3][TT + 1];
    int pl   = blockIdx.x;
    int pair = chunk0 + pl;
    int wrow = pair / STEPM1;
    int rcol = pair % STEPM1;
    int aseq = wrow * (STEPM1 + 1);
    int oseq = aseq + 1 + rcol;
    int la = lens[aseq], lb = lens[oseq];
    int ktarget = la + lb;
    const float* Drow = Dd + (size_t)pl * CELLS;   // base of diagonal d=0

    int t = threadIdx.x;                 // handles diagonal entry i = t+1
    buf[0][t + 1] = INFV;                // diag k=0: R[0,0]=0 else INF
    buf[1][t + 1] = INFV;                // diag k=1: all INF
    if (t == 0) { buf[0][0] = 0.0f; buf[1][0] = INFV; }
    __syncthreads();

    const float g = 5.0f, ig = 0.2f;
    int p2 = 0, p1 = 1, cu = 2;
    float result = 0.0f;

    for (int k = 2; k <= 2 * TT; ++k) {
        int d = k - 2;                   // current diagonal of D, length len(d)
        int mmin = d > 511 ? d - 511 : 0;
        float nv = INFV;
        if ((unsigned)(d - t) < (unsigned)TT) {     // col in [0,512)
            float dd = Drow[t - mmin];
            float a = buf[p2][t];        // R[i-1, j-1]
            float b = buf[p1][t];        // R[i-1, j  ]
            float c = buf[p1][t + 1];    // R[i  , j-1]
            float mn = fminf(a, fminf(b, c));
            float s  = __expf((mn - a) * ig) + __expf((mn - b) * ig)
                     + __expf((mn - c) * ig);
            nv = dd + (mn - g * __logf(s));
        }
        buf[cu][t + 1] = nv;
        if (t == 0) buf[cu][0] = INFV;
        if (k == ktarget && t + 1 == la) result = nv;   // ys[la+lb-2, la]
        __syncthreads();
        Drow += (d <= 511) ? (d + 1) : (1023 - d);      // advance to diag d+1
        int tmp = p2; p2 = p1; p1 = cu; cu = tmp;
    }
    if (t + 1 == la) {
        float laf = (float)la, lbf = (float)lb;
        dists[pair] = result / sqrtf(laf * laf + lbf * lbf);
    }
}

// ---------------------- kernel 3: triplet loss + MMD finalize --------------
__global__ void final_kernel(const float* __restrict__ dists, float* __restrict__ out) {
    __shared__ float ds[NW][STEPM1];
    __shared__ float lvinter[NW];
    __shared__ float nzr[NW];
    __shared__ float mmax[128];
    int t = threadIdx.x;
    for (int i = t; i < PAIRS; i += 128) ((float*)ds)[i] = dists[i];
    __syncthreads();

    if (t < NW) {
        const float* gd = ds[t];
        const float* nd = ds[t] + 5;
        float sum = 0.0f; int nz_s = 0, nz_r = 0;
        for (int i = 0; i < 5; ++i)
            for (int j = 0; j < 5; ++j) {
                float v = gd[i] + 0.6f - nd[j];     if (v > 0.0f) { sum += v; nz_s++; }
                float u = gd[i] + 1.0f - nd[5 + j]; if (u > 0.0f) { sum += u; nz_r++; }
            }
        float lv = sum / (float)(nz_s + nz_r + 1);
        float ca = 0.0f, cb = 0.0f;
        for (int i = 0; i < 5; ++i) { ca += gd[i]; cb += nd[i]; }
        ca *= 0.2f; cb *= 0.2f;
        float inter = fmaxf(0.0f, 0.5f - fabsf(ca - cb));
        lvinter[t] = lv + inter * 0.5f;
        nzr[t] = (float)nz_r;
    }

    float mm = -1e30f;
    if (t < 120) {                       // triu pair -> (iu, ju)
        int idx = t, row = 0, c = 15;
        while (idx >= c) { idx -= c; row++; c = 15 - row; }
        int iu = row, ju = row + 1 + idx;
        float tot[30];
        for (int i = 0; i < 15; ++i) { tot[i] = ds[iu][i]; tot[15 + i] = ds[ju][i]; }
        float L2sum = 0.0f;
        for (int i = 0; i < 30; ++i)
            for (int j = 0; j < 30; ++j) { float dl = tot[i] - tot[j]; L2sum += dl * dl; }
        float bw = L2sum / 870.0f * 0.25f;   // /(n^2-n) then / KERNEL_MUL^(5//2)
        float acc = 0.0f;
        for (int i = 0; i < 15; ++i)
            for (int j = 0; j < 15; ++j) {
                float dxx = tot[i] - tot[j];           dxx *= dxx;
                float dyy = tot[15 + i] - tot[15 + j]; dyy *= dyy;
                float dxy = tot[i] - tot[15 + j];      dxy *= dxy;
                float dyx = tot[15 + i] - tot[j];      dyx *= dyx;
                float bwc = bw, kk = 0.0f;
                for (int q = 0; q < 5; ++q) {
                    kk += expf(-dxx / bwc) + expf(-dyy / bwc)
                        - expf(-dxy / bwc) - expf(-dyx / bwc);
                    bwc *= 2.0f;
                }
                acc += kk;
            }
        mm = acc / 225.0f;
    }
    mmax[t] = mm;
    __syncthreads();

    if (t == 0) {
        float total = 0.0f, nzsum = 0.0f;
        for (int i = 0; i < NW; ++i) { total += lvinter[i]; nzsum += nzr[i]; }
        total *= (1.0f / 16.0f);
        float mx = -1e30f;
        for (int i = 0; i < 120; ++i) mx = fmaxf(mx, mmax[i]);
        float mmd1 = mx * 0.01f;
        out[0] = total + mmd1;
        out[1] = total;
        out[2] = mmd1;
        out[3] = nzsum;
    }
}

// ---------------------------------------------------------------------------
extern "C" void kernel_launch(void* const* d_in, const int* in_sizes, int n_in,
                              void* d_out, int out_size, void* d_ws, size_t ws_size,
                              hipStream_t stream) {
    (void)in_sizes; (void)n_in; (void)out_size;
    const float* data = (const float*)d_in[0];
    const int*   lens = (const int*)d_in[1];
    float* out   = (float*)d_out;
    float* ws    = (float*)d_ws;
    float* norms = ws;                               // 131072 floats
    float* dists = ws + 131072;                      // 240 floats (pad 256)
    float* Dd    = ws + 131072 + 256;                // chunk * CELLS floats

    // Adaptive chunk size (deterministic in ws_size): prefer one shot (240
    // pairs, 240 MiB of D -> maximal DP concurrency, mostly L2-resident).
    size_t head   = 131072 + 256;
    size_t availf = (ws_size / 4 > head) ? (ws_size / 4 - head) : 0;
    int chunk = (int)(availf / CELLS);
    if (chunk > PAIRS) chunk = PAIRS;
    if (chunk < 1) chunk = 1;

    norms_kernel<<<(NW * (STEPM1 + 1)) * TT / 8, 256, 0, stream>>>(data, norms);

    for (int c = 0; c < PAIRS; c += chunk) {
        int n = PAIRS - c < chunk ? PAIRS - c : chunk;
        gemm_kernel<<<n * 256, 128, 0, stream>>>(data, norms, Dd, c);
        dtw_kernel<<<n, TT, 0, stream>>>(Dd, lens, dists, c);
    }

    final_kernel<<<1, 128, 0, stream>>>(dists, out);
}